// AttMatrixCov_61297773249009
// MI455X (gfx1250) — compile-verified
//
#include <hip/hip_runtime.h>

typedef __attribute__((ext_vector_type(2))) float v2f;
typedef __attribute__((ext_vector_type(8))) float v8f;

#define NATT   4
#define NTEMP  8
#define C_DIM  1024
#define HW     256
#define NPAIR  28

// triu_indices(8, k=1)
__constant__ int I_IDX[NPAIR] = {0,0,0,0,0,0,0,1,1,1,1,1,1,2,2,2,2,2,3,3,3,3,4,4,4,5,5,6};
__constant__ int J_IDX[NPAIR] = {1,2,3,4,5,6,7,2,3,4,5,6,7,3,4,5,6,7,4,5,6,7,5,6,7,6,7,7};

__global__ void zero_out_kernel(float* out) {
    if (threadIdx.x == 0) out[0] = 0.0f;
}

// loss_c: closed form  (Σx²)(Σy²) − 2(x·y) + C, divided by C².  One wave per pair.
__global__ __launch_bounds__(128) void loss_c_kernel(const float* __restrict__ attc,
                                                     float* __restrict__ out) {
    int wid  = blockIdx.x * 4 + (threadIdx.x >> 5);   // 0..111
    int lane = threadIdx.x & 31;
    int a = wid / NPAIR, p = wid % NPAIR;
    const float* x = attc + (size_t)(a * NTEMP + I_IDX[p]) * C_DIM;
    const float* y = attc + (size_t)(a * NTEMP + J_IDX[p]) * C_DIM;
    float sx = 0.f, sy = 0.f, dxy = 0.f;
    for (int t = lane; t < C_DIM; t += 32) {
        float xv = x[t], yv = y[t];
        sx += xv * xv; sy += yv * yv; dxy += xv * yv;
    }
    for (int off = 16; off; off >>= 1) {
        sx  += __shfl_down(sx,  off, 32);
        sy  += __shfl_down(sy,  off, 32);
        dxy += __shfl_down(dxy, off, 32);
    }
    if (lane == 0) {
        float contrib = (sx * sy - 2.0f * dxy + (float)C_DIM)
                        * (1.0f / ((float)C_DIM * (float)C_DIM));
        atomicAdd(out, contrib);
    }
}

// loss_s: G = S1^T S2 per pair via v_wmma_f32_16x16x4_f32, fused (G-I)^2 reduction.
// One 256-thread WG computes a 64x64 tile of G; 8 waves x 2 D-tiles each.
#define BLK     64
#define KSLAB   32
#define LSTRIDE 80   // 80*2 = 160 ≡ 32 (mod 64 banks): lane halves hit disjoint banks

__global__ __launch_bounds__(256) void loss_s_kernel(const float* __restrict__ atts,
                                                     float* __restrict__ out) {
    __shared__ float lds1[KSLAB * LSTRIDE];
    __shared__ float lds2[KSLAB * LSTRIDE];

    const int bpp   = (HW / BLK) * (HW / BLK);          // 16 tiles per pair
    int pair  = blockIdx.x / bpp;
    int bip   = blockIdx.x % bpp;
    int c_blk = (bip / (HW / BLK)) * BLK;
    int d_blk = (bip % (HW / BLK)) * BLK;
    int a = pair / NPAIR, p = pair % NPAIR;
    const float* S1 = atts + (size_t)(a * NTEMP + I_IDX[p]) * HW * HW;
    const float* S2 = atts + (size_t)(a * NTEMP + J_IDX[p]) * HW * HW;

    int tid   = threadIdx.x;
    int wave  = tid >> 5;                // 0..7
    int lane  = tid & 31;
    int c_sub  = (wave & 3) * 16;        // 0..48 within 64-wide block (M)
    int d_sub0 = (wave >> 2) * 32;       // 0 or 32 (two N tiles per wave)
    int m  = lane & 15;
    int kb = (lane < 16) ? 0 : 2;        // fp32 A/B fragment K split

    v8f acc0 = {}; v8f acc1 = {};

    int r  = tid >> 3;                   // 0..31  (row within K slab)
    int cq = (tid & 7) * 8;              // 0..56  (8-float chunk within 64 cols)

    for (int h0 = 0; h0 < HW; h0 += KSLAB) {
        __syncthreads();
        const float4* g1 = (const float4*)(S1 + (size_t)(h0 + r) * HW + c_blk + cq);
        const float4* g2 = (const float4*)(S2 + (size_t)(h0 + r) * HW + d_blk + cq);
        float4 x0 = g1[0], x1 = g1[1];
        float4 y0 = g2[0], y1 = g2[1];
        *(float4*)&lds1[r * LSTRIDE + cq]     = x0;
        *(float4*)&lds1[r * LSTRIDE + cq + 4] = x1;
        *(float4*)&lds2[r * LSTRIDE + cq]     = y0;
        *(float4*)&lds2[r * LSTRIDE + cq + 4] = y1;
        __syncthreads();

        #pragma unroll
        for (int kk = 0; kk < KSLAB; kk += 4) {
            v2f av, b0, b1;
            av[0] = lds1[(kk + kb    ) * LSTRIDE + c_sub + m];
            av[1] = lds1[(kk + kb + 1) * LSTRIDE + c_sub + m];
            b0[0] = lds2[(kk + kb    ) * LSTRIDE + d_sub0 + m];
            b0[1] = lds2[(kk + kb + 1) * LSTRIDE + d_sub0 + m];
            b1[0] = lds2[(kk + kb    ) * LSTRIDE + d_sub0 + 16 + m];
            b1[1] = lds2[(kk + kb + 1) * LSTRIDE + d_sub0 + 16 + m];
            acc0 = __builtin_amdgcn_wmma_f32_16x16x4_f32(
                       false, av, false, b0, (short)0, acc0, false, false);
            acc1 = __builtin_amdgcn_wmma_f32_16x16x4_f32(
                       false, av, false, b1, (short)0, acc1, false, false);
        }
    }

    // Fused (G - I)^2 on accumulator fragments.
    // C/D layout: VGPR v -> M = v (lanes 0-15) / v+8 (lanes 16-31), N = lane%16.
    float local = 0.f;
    int rb = (lane < 16) ? 0 : 8;
    int n  = lane & 15;
    int cg = c_blk + c_sub + rb;
    #pragma unroll
    for (int v = 0; v < 8; ++v) {
        int cc  = cg + v;
        int dd0 = d_blk + d_sub0 + n;
        int dd1 = dd0 + 16;
        float g0 = acc0[v] - ((cc == dd0) ? 1.0f : 0.0f);
        float g1 = acc1[v] - ((cc == dd1) ? 1.0f : 0.0f);
        local += g0 * g0 + g1 * g1;
    }
    for (int off = 16; off; off >>= 1) local += __shfl_down(local, off, 32);
    if (lane == 0) atomicAdd(out, local * (1.0f / (float)(HW * HW)));
}

extern "C" void kernel_launch(void* const* d_in, const int* in_sizes, int n_in,
                              void* d_out, int out_size, void* d_ws, size_t ws_size,
                              hipStream_t stream) {
    const float* attc = (const float*)d_in[0];
    const float* atts = (const float*)d_in[1];
    float* out = (float*)d_out;

    zero_out_kernel<<<1, 32, 0, stream>>>(out);
    loss_c_kernel<<<(NATT * NPAIR) / 4, 128, 0, stream>>>(attc, out);
    loss_s_kernel<<<NATT * NPAIR * 16, 256, 0, stream>>>(atts, out);
}